// Model_36292473652083
// MI455X (gfx1250) — compile-verified
//
#include <hip/hip_runtime.h>
#include <math.h>

// ---------------------------------------------------------------------------
// Sizes from the reference
// ---------------------------------------------------------------------------
#define NN     8192
#define EE     262144
#define D_IN   256
#define D_LAT  128
#define DROP_P 0.3f
#define NEG_SLOPE 0.2f
#define EPS_BN 1e-5f
#define INV_TEMP 2.0f   // 1 / 0.5

typedef __attribute__((ext_vector_type(2))) float v2f;
typedef __attribute__((ext_vector_type(8))) float v8f;

// ---------------------------------------------------------------------------
// Helpers
// ---------------------------------------------------------------------------
__device__ __forceinline__ unsigned f2ord(float f) {
    unsigned u = __float_as_uint(f);
    return (u & 0x80000000u) ? ~u : (u | 0x80000000u);
}
__device__ __forceinline__ float ord2f(unsigned u) {
    return (u & 0x80000000u) ? __uint_as_float(u & 0x7fffffffu)
                             : __uint_as_float(~u);
}
#define ORD_NEG_INF 0x007FFFFFu   // f2ord(-inf)

__device__ __forceinline__ unsigned hash_u32(unsigned x) {
    x ^= x >> 16; x *= 0x7feb352dU;
    x ^= x >> 15; x *= 0x846ca68bU;
    x ^= x >> 16; return x;
}
// deterministic bijection on [0, 8192)
__device__ __forceinline__ int permute13(int i) {
    unsigned j = (unsigned)i;
    j = (j * 2741u + 1337u) & 8191u;
    j ^= j >> 7;
    j = (j * 3669u + 421u) & 8191u;
    j ^= j >> 5;
    return (int)(j & 8191u);
}

// ---------------------------------------------------------------------------
// Misc small kernels
// ---------------------------------------------------------------------------
__global__ void zero_kernel(float* p, int n) {
    int i = blockIdx.x * blockDim.x + threadIdx.x;
    if (i < n) p[i] = 0.0f;
}

__global__ void make_xp_xn(const float* __restrict__ x,
                           float* __restrict__ xp, float* __restrict__ xn,
                           int rows, int D) {
    int idx = blockIdx.x * blockDim.x + threadIdx.x;
    if (idx >= rows * D) return;
    int r = idx / D, c = idx - r * D;
    unsigned h = hash_u32(0xC0FFEEu ^ (unsigned)c);
    float u = (float)(h >> 8) * (1.0f / 16777216.0f);
    float keep = (u >= DROP_P) ? 1.0f : 0.0f;
    xp[idx] = x[idx] * keep;
    xn[idx] = x[permute13(r) * D + c];
}

// out = bias broadcast; mord = ord(-inf); denom = 0
__global__ void gat_init(float* __restrict__ out, const float* __restrict__ bias,
                         unsigned* __restrict__ mord, float* __restrict__ den,
                         int rows, int D) {
    int idx = blockIdx.x * blockDim.x + threadIdx.x;
    if (idx < rows * D) out[idx] = bias[idx % D];
    if (idx < rows) { mord[idx] = ORD_NEG_INF; den[idx] = 0.0f; }
}

// ---------------------------------------------------------------------------
// Dual GEMM with FP32 WMMA:  C1 = A @ B1, C2 = A @ B2
// A [M,K] rowmajor, B [K,N] rowmajor. grid = M/16, block = 256 (8 waves).
// Each wave owns column tiles nt = wave, wave+8, ...
// ---------------------------------------------------------------------------
__global__ __launch_bounds__(256) void gemm_dual_wmma(
    const float* __restrict__ A, const float* __restrict__ B1,
    const float* __restrict__ B2, float* __restrict__ C1, float* __restrict__ C2,
    int M, int K, int N) {
    const int lane = threadIdx.x & 31;
    const int wave = threadIdx.x >> 5;
    const int half = lane >> 4;       // 0 / 1
    const int l16  = lane & 15;
    const int row  = blockIdx.x * 16 + l16;   // A-fragment row (M = lane%16)
    const int ntiles = N >> 4;
    for (int nt = wave; nt < ntiles; nt += 8) {
        const int col = nt * 16 + l16;
        v8f c1 = {0.f,0.f,0.f,0.f,0.f,0.f,0.f,0.f};
        v8f c2 = {0.f,0.f,0.f,0.f,0.f,0.f,0.f,0.f};
        for (int k0 = 0; k0 < K; k0 += 4) {
            const int ka = k0 + half * 2;     // per-lane K pair
            v2f a = { A[row * K + ka], A[row * K + ka + 1] };
            v2f b = { B1[ka * N + col], B1[(ka + 1) * N + col] };
            c1 = __builtin_amdgcn_wmma_f32_16x16x4_f32(false, a, false, b,
                                                       (short)0, c1, false, false);
            v2f d = { B2[ka * N + col], B2[(ka + 1) * N + col] };
            c2 = __builtin_amdgcn_wmma_f32_16x16x4_f32(false, a, false, d,
                                                       (short)0, c2, false, false);
        }
#pragma unroll
        for (int v = 0; v < 8; ++v) {         // C: row = v + 8*half
            int r = blockIdx.x * 16 + v + 8 * half;
            C1[r * N + col] = c1[v];
            C2[r * N + col] = c2[v];
        }
    }
}

// ---------------------------------------------------------------------------
// GAT edge kernels (wave per edge)
// ---------------------------------------------------------------------------
__global__ __launch_bounds__(256) void edge_logit_kernel(
    const float* __restrict__ xl, const float* __restrict__ xr,
    const float* __restrict__ att, const int* __restrict__ src,
    const int* __restrict__ dst, float* __restrict__ logit,
    unsigned* __restrict__ mord, int E, int D) {
    int e = blockIdx.x * 8 + (threadIdx.x >> 5);
    if (e >= E) return;
    int lane = threadIdx.x & 31;
    int s = src[e], d = dst[e];
    float acc = 0.0f;
    for (int i = lane; i < D; i += 32) {
        float v = xl[s * D + i] + xr[d * D + i];
        v = (v > 0.0f) ? v : NEG_SLOPE * v;
        acc += v * att[i];
    }
#pragma unroll
    for (int off = 16; off; off >>= 1) acc += __shfl_down(acc, off, 32);
    if (lane == 0) {
        logit[e] = acc;
        atomicMax(mord + d, f2ord(acc));
    }
}

__global__ void m_fix_kernel(const unsigned* __restrict__ mord,
                             float* __restrict__ m, int n) {
    int i = blockIdx.x * blockDim.x + threadIdx.x;
    if (i >= n) return;
    float v = ord2f(mord[i]);
    if (isinf(v) || isnan(v)) v = 0.0f;   // nodes with no in-edges
    m[i] = v;
}

__global__ void edge_exp_kernel(float* __restrict__ logit,
                                const float* __restrict__ m,
                                const int* __restrict__ dst,
                                float* __restrict__ den, int E) {
    int e = blockIdx.x * blockDim.x + threadIdx.x;
    if (e >= E) return;
    float ex = expf(logit[e] - m[dst[e]]);
    logit[e] = ex;
    atomicAdd(den + dst[e], ex);
}

__global__ __launch_bounds__(256) void edge_scatter_kernel(
    const float* __restrict__ xl, const float* __restrict__ ex,
    const float* __restrict__ den, const float* __restrict__ ew,
    const int* __restrict__ src, const int* __restrict__ dst,
    float* __restrict__ out, int E, int D) {
    int e = blockIdx.x * 8 + (threadIdx.x >> 5);
    if (e >= E) return;
    int lane = threadIdx.x & 31;
    int s = src[e], d = dst[e];
    float alpha = ex[e] / (den[d] + 1e-16f) * ew[e];
    for (int i = lane; i < D; i += 32)
        atomicAdd(out + d * D + i, alpha * xl[s * D + i]);
}

// ---------------------------------------------------------------------------
// BatchNorm (training mode, biased var) + ELU
// ---------------------------------------------------------------------------
__global__ __launch_bounds__(128) void bn_stats(const float* __restrict__ t,
                                                float* __restrict__ sum,
                                                float* __restrict__ sumsq,
                                                int rows, int D) {
    int c = threadIdx.x;             // blockDim == D == 128
    int r0 = blockIdx.x * 64;
    float a = 0.0f, b = 0.0f;
#pragma unroll 4
    for (int i = 0; i < 64; ++i) {
        float v = t[(r0 + i) * D + c];
        a += v; b += v * v;
    }
    atomicAdd(sum + c, a);
    atomicAdd(sumsq + c, b);
}

__global__ void bn_apply_elu(const float* __restrict__ t,
                             const float* __restrict__ sum,
                             const float* __restrict__ sumsq,
                             const float* __restrict__ gamma,
                             const float* __restrict__ beta,
                             float* __restrict__ out, int rows, int D) {
    int idx = blockIdx.x * blockDim.x + threadIdx.x;
    if (idx >= rows * D) return;
    int c = idx % D;
    float inv_n = 1.0f / (float)rows;
    float mu  = sum[c] * inv_n;
    float var = sumsq[c] * inv_n - mu * mu;
    float y = (t[idx] - mu) * rsqrtf(var + EPS_BN) * gamma[c] + beta[c];
    out[idx] = (y > 0.0f) ? y : (expf(y) - 1.0f);
}

// ---------------------------------------------------------------------------
// Row L2 normalize (in place), row dot product
// ---------------------------------------------------------------------------
__global__ __launch_bounds__(256) void l2norm_kernel(float* __restrict__ v,
                                                     int rows, int D) {
    int row = blockIdx.x * 8 + (threadIdx.x >> 5);
    if (row >= rows) return;
    int lane = threadIdx.x & 31;
    float a = 0.0f;
    for (int i = lane; i < D; i += 32) { float t = v[row * D + i]; a += t * t; }
#pragma unroll
    for (int off = 16; off; off >>= 1) a += __shfl_xor(a, off, 32);
    float inv = rsqrtf(a);
    for (int i = lane; i < D; i += 32) v[row * D + i] *= inv;
}

__global__ __launch_bounds__(256) void rowdot_kernel(const float* __restrict__ a,
                                                     const float* __restrict__ b,
                                                     float* __restrict__ outv,
                                                     int rows, int D) {
    int row = blockIdx.x * 8 + (threadIdx.x >> 5);
    if (row >= rows) return;
    int lane = threadIdx.x & 31;
    float acc = 0.0f;
    for (int i = lane; i < D; i += 32) acc += a[row * D + i] * b[row * D + i];
#pragma unroll
    for (int off = 16; off; off >>= 1) acc += __shfl_down(acc, off, 32);
    if (lane == 0) outv[row] = acc;
}

// ---------------------------------------------------------------------------
// MSE reduction: acc += sum((x-h)^2)
// ---------------------------------------------------------------------------
__global__ void mse_kernel(const float* __restrict__ x, const float* __restrict__ h,
                           float* __restrict__ acc, int n) {
    int tid = blockIdx.x * blockDim.x + threadIdx.x;
    float a = 0.0f;
    for (int i = tid; i < n; i += gridDim.x * blockDim.x) {
        float d = x[i] - h[i];
        a += d * d;
    }
#pragma unroll
    for (int off = 16; off; off >>= 1) a += __shfl_down(a, off, 32);
    if ((threadIdx.x & 31) == 0) atomicAdd(acc, a);
}

// ---------------------------------------------------------------------------
// InfoNCE: per 16-row block, stream all 8192 nk columns with WMMA + online
// logsumexp. Q, NK already L2-normalized; POS = per-row positive dot.
// grid = NN/16, block = 256 (8 waves, each takes every 8th column tile).
// ---------------------------------------------------------------------------
__global__ __launch_bounds__(256) void infonce_kernel(
    const float* __restrict__ Q, const float* __restrict__ NK,
    const float* __restrict__ POS, float* __restrict__ lossAcc, int rows) {
    __shared__ float ls[8 * 16 * 17];
    __shared__ float sm[8][16];
    __shared__ float ss[8][16];
    const int lane = threadIdx.x & 31;
    const int wave = threadIdx.x >> 5;
    const int half = lane >> 4;
    const int l16  = lane & 15;
    const int row0 = blockIdx.x * 16;

    // Preload the q fragments for this 16-row tile (held in VGPRs, reused for
    // every column tile): lane holds A[l16][4t + 2*half + j].
    float aq[64];
#pragma unroll
    for (int t = 0; t < 32; ++t) {
        int k = t * 4 + half * 2;
        aq[2 * t]     = Q[(row0 + l16) * D_LAT + k];
        aq[2 * t + 1] = Q[(row0 + l16) * D_LAT + k + 1];
    }

    float m = -3.0e38f, s = 0.0f;
    float* tile = ls + wave * 272;
    const int ntiles = rows >> 4;
    for (int ct = wave; ct < ntiles; ct += 8) {
        const int col = ct * 16 + l16;
        v8f c = {0.f,0.f,0.f,0.f,0.f,0.f,0.f,0.f};
#pragma unroll
        for (int t = 0; t < 32; ++t) {
            int k = t * 4 + half * 2;
            v2f a = { aq[2 * t], aq[2 * t + 1] };
            // B[k][n] = nk[n][k]  (nk.T)
            v2f b = { NK[col * D_LAT + k], NK[col * D_LAT + k + 1] };
            c = __builtin_amdgcn_wmma_f32_16x16x4_f32(false, a, false, b,
                                                      (short)0, c, false, false);
        }
        // spill the 16x16 score tile to this wave's LDS slab
#pragma unroll
        for (int v = 0; v < 8; ++v)
            tile[(v + 8 * half) * 17 + l16] = c[v];
        asm volatile("s_wait_dscnt 0x0" ::: "memory");   // wave-local LDS sync
        if (lane < 16) {                                  // lane r owns row r
            float vals[16];
            float tmax = -3.0e38f;
#pragma unroll
            for (int j = 0; j < 16; ++j) {
                float v = tile[lane * 17 + j] * INV_TEMP;
                vals[j] = v;
                tmax = fmaxf(tmax, v);
            }
            float mn = fmaxf(m, tmax);
            float acc = 0.0f;
#pragma unroll
            for (int j = 0; j < 16; ++j) acc += __expf(vals[j] - mn);
            s = s * __expf(m - mn) + acc;
            m = mn;
        }
    }
    if (lane < 16) { sm[wave][lane] = m; ss[wave][lane] = s; }
    __syncthreads();
    if (wave == 0 && lane < 16) {
        float M = -3.0e38f;
#pragma unroll
        for (int w = 0; w < 8; ++w) M = fmaxf(M, sm[w][lane]);
        float S = 0.0f;
#pragma unroll
        for (int w = 0; w < 8; ++w) S += ss[w][lane] * __expf(sm[w][lane] - M);
        float p  = POS[row0 + lane] * INV_TEMP;
        float M2 = fmaxf(M, p);
        float S2 = S * __expf(M - M2) + __expf(p - M2);
        float lse = M2 + __logf(S2);
        atomicAdd(lossAcc, lse - p);
    }
}

__global__ void finalize_kernel(const float* __restrict__ acc,
                                float* __restrict__ out) {
    out[0] = acc[0] / ((float)NN * (float)D_IN) + 0.2f * (acc[1] / (float)NN);
}

// ---------------------------------------------------------------------------
// Host-side orchestration
// ---------------------------------------------------------------------------
extern "C" void kernel_launch(void* const* d_in, const int* in_sizes, int n_in,
                              void* d_out, int out_size, void* d_ws, size_t ws_size,
                              hipStream_t stream) {
    (void)in_sizes; (void)n_in; (void)out_size;

    const float* x      = (const float*)d_in[0];
    const int*   g_s    = (const int*)d_in[1];
    const int*   g_f    = (const int*)d_in[2];
    const float* w_h    = (const float*)d_in[3];
    const float* w_h_a  = (const float*)d_in[4];
    const float* enc_Wl = (const float*)d_in[5];
    const float* enc_Wr = (const float*)d_in[6];
    const float* enc_att= (const float*)d_in[7];
    const float* enc_b  = (const float*)d_in[8];
    const float* dec_Wl = (const float*)d_in[9];
    const float* dec_Wr = (const float*)d_in[10];
    const float* dec_att= (const float*)d_in[11];
    const float* dec_b  = (const float*)d_in[12];
    const float* bn_g   = (const float*)d_in[13];
    const float* bn_b   = (const float*)d_in[14];

    // outputs: [hi (NN*D_LAT) | h (NN*D_IN) | loss (1)]
    float* hi = (float*)d_out;
    float* h  = hi + (size_t)NN * D_LAT;
    float* lossOut = hi + (size_t)NN * D_LAT + (size_t)NN * D_IN;

    // workspace layout (floats)
    float* ws = (float*)d_ws;
    size_t off = 0;
    float* XL   = ws + off; off += (size_t)NN * D_IN;    // max(Dout) = 256
    float* XR   = ws + off; off += (size_t)NN * D_IN;
    float* XP   = ws + off; off += (size_t)NN * D_IN;
    float* XN   = ws + off; off += (size_t)NN * D_IN;
    float* H0   = ws + off; off += (size_t)NN * D_LAT;
    float* H1   = ws + off; off += (size_t)NN * D_LAT;
    float* H2   = ws + off; off += (size_t)NN * D_LAT;
    float* EDGE = ws + off; off += (size_t)EE;
    float* Mv   = ws + off; off += (size_t)NN;           // seg-max (ord uint / float)
    float* DEN  = ws + off; off += (size_t)NN;
    float* POS  = ws + off; off += (size_t)NN;
    float* STAT = ws + off; off += 512;                  // [sum(128) | sumsq(128)]
    float* LOSS = ws + off; off += 8;                    // [rec_sum, ctr_sum]
    if (ws_size < off * sizeof(float)) return;

    auto run_gat = [&](const float* xin, int Din, int Dout,
                       const float* Wl, const float* Wr,
                       const float* att, const float* bias,
                       const int* ei, const float* ew, float* outp) {
        gemm_dual_wmma<<<NN / 16, 256, 0, stream>>>(xin, Wl, Wr, XL, XR,
                                                    NN, Din, Dout);
        int tot = NN * Dout;
        gat_init<<<(tot + 255) / 256, 256, 0, stream>>>(outp, bias,
                                                        (unsigned*)Mv, DEN, NN, Dout);
        edge_logit_kernel<<<EE / 8, 256, 0, stream>>>(XL, XR, att, ei, ei + EE,
                                                      EDGE, (unsigned*)Mv, EE, Dout);
        m_fix_kernel<<<NN / 256, 256, 0, stream>>>((unsigned*)Mv, Mv, NN);
        edge_exp_kernel<<<EE / 256, 256, 0, stream>>>(EDGE, Mv, ei + EE, DEN, EE);
        edge_scatter_kernel<<<EE / 8, 256, 0, stream>>>(XL, EDGE, DEN, ew,
                                                        ei, ei + EE, outp, EE, Dout);
    };

    auto run_bn = [&](const float* tin, float* tout) {
        zero_kernel<<<1, 256, 0, stream>>>(STAT, 256);
        bn_stats<<<NN / 64, 128, 0, stream>>>(tin, STAT, STAT + 128, NN, D_LAT);
        bn_apply_elu<<<(NN * D_LAT + 255) / 256, 256, 0, stream>>>(
            tin, STAT, STAT + 128, bn_g, bn_b, tout, NN, D_LAT);
    };

    zero_kernel<<<1, 64, 0, stream>>>(LOSS, 8);
    make_xp_xn<<<(NN * D_IN + 255) / 256, 256, 0, stream>>>(x, XP, XN, NN, D_IN);

    // hi = enc(x, g_s, w_h)
    run_gat(x, D_IN, D_LAT, enc_Wl, enc_Wr, enc_att, enc_b, g_s, w_h, hi);
    // h0 = elu(bn(hi))
    run_bn(hi, H0);
    // h = dec(hi, g_s, w_h)
    run_gat(hi, D_LAT, D_IN, dec_Wl, dec_Wr, dec_att, dec_b, g_s, w_h, h);
    // loss_rec partial sum
    mse_kernel<<<512, 256, 0, stream>>>(x, h, LOSS, NN * D_IN);
    // h1 = elu(bn(enc(x_p, g_f, w_h_a)))
    run_gat(XP, D_IN, D_LAT, enc_Wl, enc_Wr, enc_att, enc_b, g_f, w_h_a, H1);
    run_bn(H1, H1);
    // h2 = elu(bn(enc(x_n, g_s, w_h)))
    run_gat(XN, D_IN, D_LAT, enc_Wl, enc_Wr, enc_att, enc_b, g_s, w_h, H2);
    run_bn(H2, H2);

    // InfoNCE
    l2norm_kernel<<<NN / 8, 256, 0, stream>>>(H0, NN, D_LAT);
    l2norm_kernel<<<NN / 8, 256, 0, stream>>>(H1, NN, D_LAT);
    l2norm_kernel<<<NN / 8, 256, 0, stream>>>(H2, NN, D_LAT);
    rowdot_kernel<<<NN / 8, 256, 0, stream>>>(H0, H1, POS, NN, D_LAT);
    infonce_kernel<<<NN / 16, 256, 0, stream>>>(H0, H2, POS, LOSS + 1, NN);

    finalize_kernel<<<1, 1, 0, stream>>>(LOSS, lossOut);
}